// TemporalGNN_72404558676832
// MI455X (gfx1250) — compile-verified
//
#include <hip/hip_runtime.h>
#include <cstdint>
#include <cmath>

// Problem constants (match reference)
#define NNODES 50000
#define DHID   128
#define TSTEPS 8
#define NEDGE  600000

typedef float v2f __attribute__((ext_vector_type(2)));
typedef float v8f __attribute__((ext_vector_type(8)));

// ---------------------------------------------------------------------------
// Fused GEMM + bias + ReLU using V_WMMA_F32_16X16X4_F32.
//   out[m,n] = relu( sum_k A0[m,k]*invdeg(m)*W0[n,k]
//                  + sum_k A1[m,k]*W1[n,k]  + bias[n] )
// A1/W1 optional (pre-layer pass). deg optional (no scaling).
// Block = 256 threads = 8 waves; each wave computes a 16x128 output tile.
// Weights are staged transposed in 64KB static LDS (Bs[k][n] = W[n][k]),
// one 128x128 matrix at a time (two phases for the two-GEMM fusion).
// ---------------------------------------------------------------------------
__global__ __launch_bounds__(256) void fused_gemm_relu(
    const float* __restrict__ A0, const float* __restrict__ deg,
    const float* __restrict__ W0,
    const float* __restrict__ A1, const float* __restrict__ W1,
    const float* __restrict__ bias,
    float* __restrict__ out, int nrows)
{
  __shared__ float Bs[DHID * DHID];  // 64 KB

  const int tid  = threadIdx.x;
  const int lane = tid & 31;
  const int wave = tid >> 5;
  const int g    = lane >> 4;   // lane-half select (K sub-offset)
  const int l16  = lane & 15;
  const int m0   = blockIdx.x * 128 + wave * 16;
  const int mrow = m0 + l16;
  const int mclamp = (mrow < nrows) ? mrow : (nrows - 1);

  v8f zero = {};
  v8f acc[8];
#pragma unroll
  for (int i = 0; i < 8; ++i) acc[i] = zero;

  // ---- phase 0: W0 / A0 (optionally scaled by 1/max(deg,1)) ----
  for (int idx = tid; idx < DHID * DHID; idx += 256) {
    int n = idx >> 7, k = idx & 127;
    Bs[(k << 7) | n] = W0[idx];          // Bs[k][n] = W0[n][k]
  }
  __syncthreads();
  {
    const float* arow = A0 + (size_t)mclamp * DHID;
    float invd = 1.0f;
    if (deg) invd = 1.0f / fmaxf(deg[mclamp], 1.0f);
#pragma unroll 1
    for (int kk = 0; kk < 32; ++kk) {
      const int k = kk * 4 + 2 * g;      // A 16x4 layout: K = kb + 2g + v
      v2f a; a.x = arow[k] * invd; a.y = arow[k + 1] * invd;
      const float* bp = &Bs[(k << 7) + l16];
#pragma unroll
      for (int nb = 0; nb < 8; ++nb) {
        v2f b; b.x = bp[nb * 16]; b.y = bp[128 + nb * 16]; // B row = kb+2g+v
        acc[nb] = __builtin_amdgcn_wmma_f32_16x16x4_f32(
            false, a, false, b, (short)0, acc[nb], false, false);
      }
    }
  }

  // ---- phase 1: W1 / A1 (skipped for the pre-layer) ----
  if (A1) {
    __syncthreads();
    for (int idx = tid; idx < DHID * DHID; idx += 256) {
      int n = idx >> 7, k = idx & 127;
      Bs[(k << 7) | n] = W1[idx];
    }
    __syncthreads();
    const float* arow = A1 + (size_t)mclamp * DHID;
#pragma unroll 1
    for (int kk = 0; kk < 32; ++kk) {
      const int k = kk * 4 + 2 * g;
      v2f a; a.x = arow[k]; a.y = arow[k + 1];
      const float* bp = &Bs[(k << 7) + l16];
#pragma unroll
      for (int nb = 0; nb < 8; ++nb) {
        v2f b; b.x = bp[nb * 16]; b.y = bp[128 + nb * 16];
        acc[nb] = __builtin_amdgcn_wmma_f32_16x16x4_f32(
            false, a, false, b, (short)0, acc[nb], false, false);
      }
    }
  }

  // ---- epilogue: bias + relu, C/D layout (VGPR r, lane-half g -> M=r+8g) ----
#pragma unroll
  for (int nb = 0; nb < 8; ++nb) {
    const int n = nb * 16 + l16;
    const float bv = bias[n];
#pragma unroll
    for (int r = 0; r < 8; ++r) {
      const int m = m0 + r + 8 * g;
      if (m < nrows) {
        float v = acc[nb][r] + bv;
        out[(size_t)m * DHID + n] = v > 0.0f ? v : 0.0f;
      }
    }
  }
}

// ---------------------------------------------------------------------------
// One wave per edge: agg[dst,:] += h[src,:], deg[dst] += 1 (float atomics).
// agg (25.6 MB) fits in the 192 MB L2, so atomics are L2-side.
// ---------------------------------------------------------------------------
__global__ __launch_bounds__(256) void scatter_mean_accum(
    const float* __restrict__ h, const int* __restrict__ edges_t,
    float* __restrict__ agg, float* __restrict__ deg, int nedges)
{
  const int gw   = (int)((blockIdx.x * (size_t)blockDim.x + threadIdx.x) >> 5);
  const int lane = threadIdx.x & 31;
  if (gw >= nedges) return;
  const int src = edges_t[gw];
  const int dst = edges_t[(size_t)nedges + gw];
  const float4 v = ((const float4*)(h + (size_t)src * DHID))[lane];
  float* ap = agg + (size_t)dst * DHID + lane * 4;
  atomicAdd(ap + 0, v.x);
  atomicAdd(ap + 1, v.y);
  atomicAdd(ap + 2, v.z);
  atomicAdd(ap + 3, v.w);
  if (lane == 0) atomicAdd(&deg[dst], 1.0f);
}

__global__ void zero_k(float* __restrict__ p, size_t n) {
  size_t i = blockIdx.x * (size_t)blockDim.x + threadIdx.x;
  if (i < n) p[i] = 0.0f;
}

// One wave per node: logits[i] = dot(h[i,:], W_post) + b_post
__global__ __launch_bounds__(256) void logits_k(
    const float* __restrict__ h, const float* __restrict__ Wp,
    const float* __restrict__ bp, float* __restrict__ logits, int n)
{
  const int gw   = (int)((blockIdx.x * (size_t)blockDim.x + threadIdx.x) >> 5);
  const int lane = threadIdx.x & 31;
  if (gw >= n) return;
  const float4 hv = ((const float4*)(h + (size_t)gw * DHID))[lane];
  const float4 wv = ((const float4*)Wp)[lane];
  float s = hv.x * wv.x + hv.y * wv.y + hv.z * wv.z + hv.w * wv.w;
#pragma unroll
  for (int off = 16; off > 0; off >>= 1) s += __shfl_xor(s, off, 32);
  if (lane == 0) logits[gw] = s + bp[0];
}

// Single block: red[0] = max(logits), red[1] = log(sum exp(logits - max))
__global__ __launch_bounds__(1024) void lse_reduce_k(
    const float* __restrict__ logits, float* __restrict__ red, int n)
{
  __shared__ float sm[1024];
  const int tid = threadIdx.x;
  float m = -3.402823466e38f;
  for (int i = tid; i < n; i += 1024) m = fmaxf(m, logits[i]);
  sm[tid] = m; __syncthreads();
  for (int s = 512; s > 0; s >>= 1) {
    if (tid < s) sm[tid] = fmaxf(sm[tid], sm[tid + s]);
    __syncthreads();
  }
  const float gmax = sm[0];
  __syncthreads();
  float a = 0.0f;
  for (int i = tid; i < n; i += 1024) a += expf(logits[i] - gmax);
  sm[tid] = a; __syncthreads();
  for (int s = 512; s > 0; s >>= 1) {
    if (tid < s) sm[tid] += sm[tid + s];
    __syncthreads();
  }
  if (tid == 0) { red[0] = gmax; red[1] = logf(sm[0]); }
}

__global__ void logsoftmax_k(const float* __restrict__ logits,
                             const float* __restrict__ red,
                             float* __restrict__ out, int n)
{
  int i = blockIdx.x * blockDim.x + threadIdx.x;
  if (i < n) out[i] = logits[i] - red[0] - red[1];
}

// ---------------------------------------------------------------------------
extern "C" void kernel_launch(void* const* d_in, const int* in_sizes, int n_in,
                              void* d_out, int out_size, void* d_ws, size_t ws_size,
                              hipStream_t stream)
{
  (void)in_sizes; (void)n_in; (void)out_size; (void)ws_size;
  const float* x      = (const float*)d_in[0];
  const int*   edges  = (const int*)  d_in[1];  // [T][2][E]
  const float* W_pre  = (const float*)d_in[2];
  const float* b_pre  = (const float*)d_in[3];
  const float* Wl     = (const float*)d_in[4];  // [T][HID][HID]
  const float* bl     = (const float*)d_in[5];  // [T][HID]
  const float* Wr     = (const float*)d_in[6];  // [T][HID][HID]
  const float* W_post = (const float*)d_in[7];  // [1][HID]
  const float* b_post = (const float*)d_in[8];  // [1]
  float* out = (float*)d_out;

  float* ws = (float*)d_ws;
  const size_t NH = (size_t)NNODES * DHID;
  float* buf0   = ws;
  float* buf1   = ws + NH;
  float* deg    = ws + 2 * NH;
  float* logits = deg + NNODES;
  float* red    = logits + NNODES;

  const int gemm_blocks = (NNODES + 127) / 128;

  // h = relu(x @ W_pre.T + b_pre)
  fused_gemm_relu<<<gemm_blocks, 256, 0, stream>>>(
      x, nullptr, W_pre, nullptr, nullptr, b_pre, buf0, NNODES);

  float* hcur = buf0;
  float* hnext = buf1;
  for (int c = 0; c < TSTEPS; ++c) {
    const int t = TSTEPS - 1 - c;  // edges applied in reversed time order
    zero_k<<<(int)((NH + 255) / 256), 256, 0, stream>>>(hnext, NH);
    zero_k<<<(NNODES + 255) / 256, 256, 0, stream>>>(deg, (size_t)NNODES);
    scatter_mean_accum<<<(NEDGE * 32 + 255) / 256, 256, 0, stream>>>(
        hcur, edges + (size_t)t * 2 * NEDGE, hnext, deg, NEDGE);
    // h = relu( (agg/deg) @ Wl.T + bl + h @ Wr.T )   (writes in place to agg)
    fused_gemm_relu<<<gemm_blocks, 256, 0, stream>>>(
        hnext, deg, Wl + (size_t)t * DHID * DHID,
        hcur, Wr + (size_t)t * DHID * DHID,
        bl + (size_t)t * DHID, hnext, NNODES);
    float* tmp = hcur; hcur = hnext; hnext = tmp;
  }

  logits_k<<<(NNODES * 32 + 255) / 256, 256, 0, stream>>>(
      hcur, W_post, b_post, logits, NNODES);
  lse_reduce_k<<<1, 1024, 0, stream>>>(logits, red, NNODES);
  logsoftmax_k<<<(NNODES + 255) / 256, 256, 0, stream>>>(logits, red, out, NNODES);
}